// ClusterControl_54941221651349
// MI455X (gfx1250) — compile-verified
//
#include <hip/hip_runtime.h>
#include <hip/hip_bf16.h>
#include <math.h>

typedef float v2f __attribute__((ext_vector_type(2)));
typedef float v8f __attribute__((ext_vector_type(8)));

#define B_N   8192
#define E_K   256
#define C_N   25
#define TOPN  16              // k+1 = 16 -> threshold is the 16th-smallest distance
#define TILE  16
#define NKFRAG (E_K / 4)      // 64 wmma_f32_16x16x4 steps per 16x16 tile
#define WAVES_PER_WG 8
#define ROWS_PER_WG (WAVES_PER_WG * TILE)   // 128
#define BSTRIDE 260           // 256 + 4 pad -> conflict-free ds_load_b64 B-fragments
#define NTILES (B_N / TILE)   // 512

// CDNA5 split dependency counters (inline asm: not modeled by the builtins).
#define CC_WAIT_ASYNC() asm volatile("s_wait_asynccnt 0x0" ::: "memory")
#define CC_WAIT_DS()    asm volatile("s_wait_dscnt 0x0" ::: "memory")

// ---------------------------------------------------------------------------
// Prep: copy encodings to d_out (identity branch of the layer), row norms sq,
// and hard cluster labels = argmax(categorical) (first-max wins, like jnp).
// ---------------------------------------------------------------------------
__global__ void cc_prep_kernel(const float* __restrict__ enc,
                               const float* __restrict__ cat,
                               float* __restrict__ out_enc,
                               float* __restrict__ sq,
                               int* __restrict__ hard) {
    __shared__ float red[256];
    const int row = blockIdx.x;
    const int t = threadIdx.x;
    float v = enc[(size_t)row * E_K + t];
    out_enc[(size_t)row * E_K + t] = v;
    red[t] = v * v;
    __syncthreads();
    for (int s = 128; s > 0; s >>= 1) {
        if (t < s) red[t] += red[t + s];
        __syncthreads();
    }
    if (t == 0) {
        sq[row] = red[0];
        float best = cat[row * C_N];
        int bi = 0;
        for (int j = 1; j < C_N; ++j) {
            float c = cat[row * C_N + j];
            if (c > best) { best = c; bi = j; }
        }
        hard[row] = bi;
    }
}

// ---------------------------------------------------------------------------
// Async (ASYNCcnt-tracked) staging of one 16x256 B panel into LDS.
// Each of the 256 threads copies 64B as 4x GLOBAL_LOAD_ASYNC_TO_LDS_B128;
// the instruction offset is added to BOTH the global and the LDS address
// (ISA 10. async pseudocode), so one addr pair serves all four.
// No VGPR round-trip, and the copy overlaps the WMMA chain (double buffer).
// ---------------------------------------------------------------------------
__device__ __forceinline__ void cc_stage_async(float* __restrict__ dstbuf,
                                               const float* __restrict__ enc,
                                               int colBase, int tid) {
    const int col = tid >> 4;          // 0..15
    const int k0  = (tid & 15) * 16;   // 16 floats per thread
    const float* src = enc + (size_t)(colBase + col) * E_K + k0;
    float* dst = dstbuf + col * BSTRIDE + k0;
    const uint32_t ldsa = (uint32_t)(uintptr_t)dst;  // LDS aperture: addr[31:0]
    asm volatile(
        "global_load_async_to_lds_b128 %0, %1, off\n\t"
        "global_load_async_to_lds_b128 %0, %1, off offset:16\n\t"
        "global_load_async_to_lds_b128 %0, %1, off offset:32\n\t"
        "global_load_async_to_lds_b128 %0, %1, off offset:48"
        :: "v"(ldsa), "v"(src) : "memory");
}

// ---------------------------------------------------------------------------
// 16x16 f32 Gram tile via V_WMMA_F32_16X16X4_F32, K accumulated over 256.
// A-fragments are resident in VGPRs (reused across all 512 column tiles);
// B-fragments stream from the LDS-staged 16x256 panel.
// A layout (32-bit A 16x4): lane = {M = lane&15}, vgpr0/1 = K {0,1} (+2 if lane>=16).
// ---------------------------------------------------------------------------
__device__ __forceinline__ v8f cc_gram_tile(const v2f* __restrict__ afrag,
                                            const float* __restrict__ bcol) {
    v8f acc = {0.f, 0.f, 0.f, 0.f, 0.f, 0.f, 0.f, 0.f};
#pragma unroll
    for (int i = 0; i < NKFRAG; ++i) {
        v2f b = *(const v2f*)(bcol + 4 * i);
        acc = __builtin_amdgcn_wmma_f32_16x16x4_f32(
            false, afrag[i], false, b, (short)0, acc, false, false);
    }
    return acc;
}

// Branch-guarded bubble-insert into an ascending 16-entry smallest-list.
__device__ __forceinline__ void cc_topk_insert(float (&t)[TOPN], float d) {
    if (d < t[TOPN - 1]) {
        float v = d;
#pragma unroll
        for (int i = 0; i < TOPN; ++i) {
            float lo = fminf(v, t[i]);
            float hi = fmaxf(v, t[i]);
            t[i] = lo;
            v = hi;
        }
    }
}

// ---------------------------------------------------------------------------
// Pass 1: per-row threshold = 16th smallest Euclidean distance.
// ---------------------------------------------------------------------------
__global__ void __launch_bounds__(256, 1)
cc_thresh_kernel(const float* __restrict__ enc,
                 const float* __restrict__ sq,
                 float* __restrict__ thr) {
    __shared__ float bstage[2][TILE * BSTRIDE];
    __shared__ float dtile[WAVES_PER_WG][TILE][18];
    __shared__ float mbuf[WAVES_PER_WG][TILE][TOPN];

    const int tid  = threadIdx.x;
    const int lane = tid & 31;
    const int wave = tid >> 5;
    const int h = lane >> 4;   // lane half
    const int c = lane & 15;
    const int rowA = blockIdx.x * ROWS_PER_WG + wave * TILE;

    // A fragments for rows rowA..rowA+15, full K=256, resident in registers.
    v2f afrag[NKFRAG];
    const float* arow = enc + (size_t)(rowA + c) * E_K + 2 * h;
#pragma unroll
    for (int i = 0; i < NKFRAG; ++i) afrag[i] = *(const v2f*)(arow + 4 * i);

    float sqa[8];
#pragma unroll
    for (int r = 0; r < 8; ++r) sqa[r] = sq[rowA + r + 8 * h];

    float t[TOPN];
#pragma unroll
    for (int i = 0; i < TOPN; ++i) t[i] = 3.0e38f;

    // Prologue: stage panel 0, then pipeline: stage t+1 while computing t.
    cc_stage_async(bstage[0], enc, 0, tid);
    CC_WAIT_ASYNC();
    __syncthreads();

    for (int tl = 0; tl < NTILES; ++tl) {
        const int colBase = tl * TILE;
        const float* cb = bstage[tl & 1];
        if (tl + 1 < NTILES)
            cc_stage_async(bstage[(tl + 1) & 1], enc, colBase + TILE, tid);

        const float* bcol = cb + c * BSTRIDE + 2 * h;
        v8f acc = cc_gram_tile(afrag, bcol);
        const float sqb = sq[colBase + c];
#pragma unroll
        for (int r = 0; r < 8; ++r) {
            float d2 = sqa[r] + sqb - 2.0f * acc[r];
            dtile[wave][r + 8 * h][c] = sqrtf(fmaxf(d2, 0.0f));
        }
        // dtile write->read is same-wave cross-lane: DS ops of a wave are
        // in-order, so a dscnt wait (not a workgroup barrier) is enough.
        CC_WAIT_DS();
#pragma unroll
        for (int j = 0; j < 8; ++j)
            cc_topk_insert(t, dtile[wave][c][h * 8 + j]);

        CC_WAIT_ASYNC();   // next panel landed in the other buffer
        __syncthreads();   // all waves done reading cb before it is restaged
    }

    // Merge the two half-lane top-16 lists per row; lower half writes thr.
    if (h == 1) {
#pragma unroll
        for (int i = 0; i < TOPN; ++i) mbuf[wave][c][i] = t[i];
    }
    __syncthreads();
    if (h == 0) {
#pragma unroll
        for (int i = 0; i < TOPN; ++i) cc_topk_insert(t, mbuf[wave][c][i]);
        thr[rowA + c] = t[TOPN - 1];
    }
}

// ---------------------------------------------------------------------------
// Pass 2: recompute distances, strict mask d < thresh[row], histogram the
// neighbour cluster labels in LDS, reduce to neighbourhood entropy.
// ---------------------------------------------------------------------------
__global__ void __launch_bounds__(256, 1)
cc_entropy_kernel(const float* __restrict__ enc,
                  const float* __restrict__ sq,
                  const float* __restrict__ thr,
                  const int* __restrict__ hard,
                  float* __restrict__ out_ent) {
    __shared__ float bstage[2][TILE * BSTRIDE];
    __shared__ int counts[ROWS_PER_WG][C_N + 1];  // pad to 26 for bank spread

    const int tid  = threadIdx.x;
    const int lane = tid & 31;
    const int wave = tid >> 5;
    const int h = lane >> 4;
    const int c = lane & 15;
    const int rowA = blockIdx.x * ROWS_PER_WG + wave * TILE;

    for (int i = tid; i < ROWS_PER_WG * (C_N + 1); i += 256)
        (&counts[0][0])[i] = 0;

    v2f afrag[NKFRAG];
    const float* arow = enc + (size_t)(rowA + c) * E_K + 2 * h;
#pragma unroll
    for (int i = 0; i < NKFRAG; ++i) afrag[i] = *(const v2f*)(arow + 4 * i);

    float sqa[8], tha[8];
#pragma unroll
    for (int r = 0; r < 8; ++r) {
        sqa[r] = sq[rowA + r + 8 * h];
        tha[r] = thr[rowA + r + 8 * h];
    }

    cc_stage_async(bstage[0], enc, 0, tid);
    CC_WAIT_ASYNC();
    __syncthreads();   // counts zeroed + panel 0 visible

    for (int tl = 0; tl < NTILES; ++tl) {
        const int colBase = tl * TILE;
        const float* cb = bstage[tl & 1];
        if (tl + 1 < NTILES)
            cc_stage_async(bstage[(tl + 1) & 1], enc, colBase + TILE, tid);

        const float* bcol = cb + c * BSTRIDE + 2 * h;
        v8f acc = cc_gram_tile(afrag, bcol);
        const float sqb = sq[colBase + c];
        const int cl = hard[colBase + c];  // cluster label of this lane's column
#pragma unroll
        for (int r = 0; r < 8; ++r) {
            float d2 = sqa[r] + sqb - 2.0f * acc[r];
            float d = sqrtf(fmaxf(d2, 0.0f));
            if (d < tha[r])  // strict <, matches reference; ~16 hits / 8192
                atomicAdd(&counts[wave * TILE + r + 8 * h][cl], 1);
        }

        CC_WAIT_ASYNC();
        __syncthreads();
    }

    if (tid < ROWS_PER_WG) {
        int n = 0;
        for (int j = 0; j < C_N; ++j) n += counts[tid][j];
        const float inv = 1.0f / (float)n;
        float e = 0.0f;
        for (int j = 0; j < C_N; ++j) {
            float b = (float)counts[tid][j] * inv;
            e -= b * logf(b + 1e-5f);  // zero bins contribute exactly 0
        }
        out_ent[blockIdx.x * ROWS_PER_WG + tid] = e;
    }
}

// ---------------------------------------------------------------------------
extern "C" void kernel_launch(void* const* d_in, const int* in_sizes, int n_in,
                              void* d_out, int out_size, void* d_ws, size_t ws_size,
                              hipStream_t stream) {
    const float* enc = (const float*)d_in[0];   // [8192, 256] f32
    const float* cat = (const float*)d_in[1];   // [8192, 25]  f32
    (void)in_sizes; (void)n_in; (void)out_size; (void)ws_size;

    float* out = (float*)d_out;                 // [8192*256] enc copy + [8192] entropy
    float* sq   = (float*)d_ws;                                   // 8192 f32
    int*   hard = (int*)((char*)d_ws + (size_t)B_N * sizeof(float));    // 8192 i32
    float* thr  = (float*)((char*)d_ws + 2 * (size_t)B_N * sizeof(float)); // 8192 f32

    cc_prep_kernel<<<B_N, 256, 0, stream>>>(enc, cat, out, sq, hard);
    cc_thresh_kernel<<<B_N / ROWS_PER_WG, 256, 0, stream>>>(enc, sq, thr);
    cc_entropy_kernel<<<B_N / ROWS_PER_WG, 256, 0, stream>>>(
        enc, sq, thr, hard, out + (size_t)B_N * E_K);
}